// ConditionalDense_3075196584427
// MI455X (gfx1250) — compile-verified
//
#include <hip/hip_runtime.h>

#define DD 512
#define UU 512

typedef __attribute__((ext_vector_type(2))) float v2f;
typedef __attribute__((ext_vector_type(8))) float v8f;

// ---------------- grouping pipeline ----------------

__global__ void zero_ws_kernel(int* counts, int* cursors, int C) {
    int i = blockIdx.x * blockDim.x + threadIdx.x;
    if (i < C) { counts[i] = 0; cursors[i] = 0; }
}

__global__ void hist_kernel(const int* __restrict__ classes, int* counts, int Bn) {
    int b = blockIdx.x * blockDim.x + threadIdx.x;
    if (b < Bn) atomicAdd(&counts[classes[b]], 1);
}

__global__ void scan_kernel(const int* __restrict__ counts, int* offsets, int C) {
    if (threadIdx.x == 0 && blockIdx.x == 0) {
        int acc = 0;
        for (int c = 0; c < C; ++c) { offsets[c] = acc; acc += counts[c]; }
        offsets[C] = acc;
    }
}

__global__ void scatter_kernel(const int* __restrict__ classes,
                               const int* __restrict__ offsets,
                               int* cursors, int* rowlist, int Bn) {
    int b = blockIdx.x * blockDim.x + threadIdx.x;
    if (b < Bn) {
        int c = classes[b];
        int pos = offsets[c] + atomicAdd(&cursors[c], 1);
        rowlist[pos] = b;
    }
}

// ---------------- grouped GEMM via V_WMMA_F32_16X16X4_F32 ----------------
// grid = (C, 2). 256 threads = 8 waves; wave w owns columns [w*64, w*64+64).
// Each wave computes 16x64 output tiles (4 v8f accumulators), K-loop step 4.

__global__ void __launch_bounds__(256)
cond_dense_wmma(const float* __restrict__ x,     // [B, D]
                const float* __restrict__ wtab,  // [C, D, U]
                const float* __restrict__ bias,  // [C, U]
                const int*   __restrict__ offsets,
                const int*   __restrict__ rowlist,
                float*       __restrict__ out)   // [B, U]
{
    const int c     = blockIdx.x;
    const int start = offsets[c];
    const int cnt   = offsets[c + 1] - start;
    if (cnt == 0) return;                       // uniform per block: EXEC stays all-1s

    const int lane  = threadIdx.x & 31;
    const int wave  = threadIdx.x >> 5;         // 0..7
    const int nBase = wave * 64;
    const int laneM = lane & 15;
    const int hi    = lane >> 4;                // 0 or 1
    const int kHalf = hi * 2;                   // K sub-offset 0 or 2

    const float* __restrict__ W = wtab + (size_t)c * DD * UU;

    // bias for this lane's 4 stored columns (hoisted out of the row-tile loop)
    float bv0 = bias[(size_t)c * UU + nBase + laneM];
    float bv1 = bias[(size_t)c * UU + nBase + 16 + laneM];
    float bv2 = bias[(size_t)c * UU + nBase + 32 + laneM];
    float bv3 = bias[(size_t)c * UU + nBase + 48 + laneM];

    for (int m0 = blockIdx.y * 16; m0 < cnt; m0 += 32) {
        // A-fragment source row for this lane (clamped; invalid rows masked at store)
        int mrow   = m0 + laneM;
        int srcRow = rowlist[start + (mrow < cnt ? mrow : cnt - 1)];
        const float* __restrict__ xrow = x + (size_t)srcRow * DD;

        v8f acc0 = {}, acc1 = {}, acc2 = {}, acc3 = {};

        for (int k0 = 0; k0 < DD; k0 += 4) {
            // A: lane<16 -> (K=k0,k0+1) of row laneM; lane>=16 -> (K=k0+2,k0+3)
            const float2 a2 = *(const float2*)(xrow + k0 + kHalf);
            v2f A; A.x = a2.x; A.y = a2.y;

            // B: lanes stripe N; .x = row k0+kHalf, .y = row k0+kHalf+1
            const float* __restrict__ Wp = W + (size_t)(k0 + kHalf) * UU + nBase + laneM;
            v2f B0, B1, B2, B3;
            B0.x = Wp[0];  B0.y = Wp[UU];
            B1.x = Wp[16]; B1.y = Wp[UU + 16];
            B2.x = Wp[32]; B2.y = Wp[UU + 32];
            B3.x = Wp[48]; B3.y = Wp[UU + 48];

            acc0 = __builtin_amdgcn_wmma_f32_16x16x4_f32(false, A, false, B0, (short)0, acc0, false, false);
            acc1 = __builtin_amdgcn_wmma_f32_16x16x4_f32(false, A, false, B1, (short)0, acc1, false, false);
            acc2 = __builtin_amdgcn_wmma_f32_16x16x4_f32(false, A, false, B2, (short)0, acc2, false, false);
            acc3 = __builtin_amdgcn_wmma_f32_16x16x4_f32(false, A, false, B3, (short)0, acc3, false, false);
        }

        // D layout: VGPR r, lanes 0-15 = [M=m0+r][N=laneM], lanes 16-31 = [M=m0+8+r]
        #pragma unroll
        for (int r = 0; r < 8; ++r) {
            int m = m0 + hi * 8 + r;
            if (m < cnt) {
                int row = rowlist[start + m];
                float* o = out + (size_t)row * UU + nBase + laneM;
                o[0]  = acc0[r] + bv0;
                o[16] = acc1[r] + bv1;
                o[32] = acc2[r] + bv2;
                o[48] = acc3[r] + bv3;
            }
        }
    }
}

// ---------------- launcher ----------------

extern "C" void kernel_launch(void* const* d_in, const int* in_sizes, int n_in,
                              void* d_out, int out_size, void* d_ws, size_t ws_size,
                              hipStream_t stream) {
    const float* x       = (const float*)d_in[0];
    const int*   classes = (const int*)d_in[1];
    const float* kern    = (const float*)d_in[2];
    const float* bias    = (const float*)d_in[3];
    float*       out     = (float*)d_out;

    const int Bn = in_sizes[0] / DD;   // 2048
    const int Cn = in_sizes[3] / UU;   // 100

    int* counts  = (int*)d_ws;
    int* offsets = counts + Cn;
    int* cursors = offsets + (Cn + 1);
    int* rowlist = cursors + Cn;

    zero_ws_kernel<<<(Cn + 255) / 256, 256, 0, stream>>>(counts, cursors, Cn);
    hist_kernel<<<(Bn + 255) / 256, 256, 0, stream>>>(classes, counts, Bn);
    scan_kernel<<<1, 32, 0, stream>>>(counts, offsets, Cn);
    scatter_kernel<<<(Bn + 255) / 256, 256, 0, stream>>>(classes, offsets, cursors, rowlist, Bn);

    cond_dense_wmma<<<dim3(Cn, 2), 256, 0, stream>>>(x, kern, bias, offsets, rowlist, out);
}